// Generator_51359218925874
// MI455X (gfx1250) — compile-verified
//
#include <hip/hip_runtime.h>
#include <hip/hip_bf16.h>
#include <math.h>

#define NN 100000
#define BB 1024
#define SS 32
#define MM 64
#define ITEM_HI 39999
#define SLOPE 0.01f

typedef __attribute__((ext_vector_type(2))) float v2f;
typedef __attribute__((ext_vector_type(8))) float v8f;

static __device__ __forceinline__ v8f wmma_f32_4(v2f a, v2f b, v8f c) {
  // D = A(16x4 f32) * B(4x16 f32) + C(16x16 f32), wave32
  return __builtin_amdgcn_wmma_f32_16x16x4_f32(false, a, false, b, (short)0, c, false, false);
}

static __device__ __forceinline__ float leaky(float x) { return x >= 0.f ? x : SLOPE * x; }

static __device__ __forceinline__ unsigned hash3(unsigned a, unsigned b, unsigned c) {
  unsigned h = a * 0x9E3779B1u ^ b * 0x85EBCA77u ^ c * 0xC2B2AE3Du;
  h ^= h >> 16; h *= 0x7FEB352Du; h ^= h >> 15; h *= 0x846CA68Bu; h ^= h >> 16;
  return h;
}

// ---------------- zero / init ----------------
__global__ void zero_f32_kernel(float* __restrict__ p, long n) {
  long i = (long)blockIdx.x * blockDim.x + threadIdx.x;
  long stride = (long)gridDim.x * blockDim.x;
  for (; i < n; i += stride) p[i] = 0.f;
}

__global__ void init_p_kernel(const int* __restrict__ pos, int* __restrict__ p_cur) {
  int b = blockIdx.x * blockDim.x + threadIdx.x;
  if (b < BB) p_cur[b] = pos[b];
}

// ---------------- scatter layer 1: one wave per source node, lane = feature ----------------
__global__ void scatter1_kernel(const float* __restrict__ x, const int* __restrict__ edge,
                                float* __restrict__ sum1, float* __restrict__ cnt) {
  int wave = (blockIdx.x * blockDim.x + threadIdx.x) >> 5;
  int lane = threadIdx.x & 31;
  if (wave >= NN) return;
  float v0 = x[(size_t)wave * 64 + lane];
  float v1 = x[(size_t)wave * 64 + 32 + lane];
  int dl = edge[(size_t)wave * SS + lane];
#pragma unroll
  for (int s = 0; s < SS; ++s) {
    int d = __shfl(dl, s, 32);
    atomicAdd(&sum1[(size_t)d * 64 + lane], v0);
    atomicAdd(&sum1[(size_t)d * 64 + 32 + lane], v1);
    if (lane == 0) atomicAdd(&cnt[d], 1.0f);
  }
}

// ---------------- scatter layer 2 (32 features) ----------------
__global__ void scatter2_kernel(const float* __restrict__ h1, const int* __restrict__ edge,
                                float* __restrict__ sum2) {
  int wave = (blockIdx.x * blockDim.x + threadIdx.x) >> 5;
  int lane = threadIdx.x & 31;
  if (wave >= NN) return;
  float v = h1[(size_t)wave * 32 + lane];
  int dl = edge[(size_t)wave * SS + lane];
#pragma unroll
  for (int s = 0; s < SS; ++s) {
    int d = __shfl(dl, s, 32);
    atomicAdd(&sum2[(size_t)d * 32 + lane], v);
  }
}

// ---------------- layer 1 GEMM: h1 = leaky(mean @ W1l^T + b1 + x @ W1r^T), WMMA f32 16x16x4 ----------------
__global__ void gemm1_kernel(const float* __restrict__ sum1, const float* __restrict__ cnt,
                             const float* __restrict__ x,
                             const float* __restrict__ W1l, const float* __restrict__ W1r,
                             const float* __restrict__ bias1, float* __restrict__ h1) {
  const int NT = NN / 16;
  int wave = (blockIdx.x * blockDim.x + threadIdx.x) >> 5;
  if (wave >= NT) return;                 // wave-uniform: EXEC stays all-ones for WMMA
  int lane = threadIdx.x & 31;
  int q = lane & 15;                      // A-row-in-tile for loads; C-col for stores
  int kh = lane >> 4;                     // K-half per ISA 16x4 A layout
  int base = wave * 16;
  float invc = 1.0f / fmaxf(cnt[base + q], 1.0f);
  const float* Arow = sum1 + (size_t)(base + q) * 64;
  const float* Xrow = x + (size_t)(base + q) * 64;
  v8f acc0 = {}; v8f acc1 = {};
#pragma unroll
  for (int kk = 0; kk < 16; ++kk) {       // mean @ W1l^T, K=64
    int d0 = kk * 4 + 2 * kh;
    v2f a; a.x = Arow[d0] * invc; a.y = Arow[d0 + 1] * invc;
    v2f wb0; wb0.x = W1l[q * 64 + d0];        wb0.y = W1l[q * 64 + d0 + 1];
    v2f wb1; wb1.x = W1l[(q + 16) * 64 + d0]; wb1.y = W1l[(q + 16) * 64 + d0 + 1];
    acc0 = wmma_f32_4(a, wb0, acc0);
    acc1 = wmma_f32_4(a, wb1, acc1);
  }
#pragma unroll
  for (int kk = 0; kk < 16; ++kk) {       // x @ W1r^T, K=64
    int d0 = kk * 4 + 2 * kh;
    v2f a; a.x = Xrow[d0]; a.y = Xrow[d0 + 1];
    v2f wb0; wb0.x = W1r[q * 64 + d0];        wb0.y = W1r[q * 64 + d0 + 1];
    v2f wb1; wb1.x = W1r[(q + 16) * 64 + d0]; wb1.y = W1r[(q + 16) * 64 + d0 + 1];
    acc0 = wmma_f32_4(a, wb0, acc0);
    acc1 = wmma_f32_4(a, wb1, acc1);
  }
#pragma unroll
  for (int r = 0; r < 8; ++r) {           // C layout: row = r + 8*kh, col = q
    int row = base + r + 8 * kh;
    h1[(size_t)row * 32 + q]      = leaky(acc0[r] + bias1[q]);
    h1[(size_t)row * 32 + 16 + q] = leaky(acc1[r] + bias1[16 + q]);
  }
}

// ---------------- layer 2 GEMM + L2 row normalize: gcn ----------------
__global__ void gemm2_kernel(const float* __restrict__ sum2, const float* __restrict__ cnt,
                             const float* __restrict__ h1,
                             const float* __restrict__ W2l, const float* __restrict__ W2r,
                             const float* __restrict__ bias2, float* __restrict__ gcn) {
  const int NT = NN / 16;
  int wave = (blockIdx.x * blockDim.x + threadIdx.x) >> 5;
  if (wave >= NT) return;
  int lane = threadIdx.x & 31;
  int q = lane & 15;
  int kh = lane >> 4;
  int base = wave * 16;
  float invc = 1.0f / fmaxf(cnt[base + q], 1.0f);
  const float* Arow = sum2 + (size_t)(base + q) * 32;
  const float* Hrow = h1 + (size_t)(base + q) * 32;
  v8f acc0 = {}; v8f acc1 = {}; v8f acc2 = {}; v8f acc3 = {};
#pragma unroll
  for (int kk = 0; kk < 8; ++kk) {        // mean2 @ W2l^T, K=32
    int d0 = kk * 4 + 2 * kh;
    v2f a; a.x = Arow[d0] * invc; a.y = Arow[d0 + 1] * invc;
    v2f wb;
    wb.x = W2l[q * 32 + d0];        wb.y = W2l[q * 32 + d0 + 1];        acc0 = wmma_f32_4(a, wb, acc0);
    wb.x = W2l[(q + 16) * 32 + d0]; wb.y = W2l[(q + 16) * 32 + d0 + 1]; acc1 = wmma_f32_4(a, wb, acc1);
    wb.x = W2l[(q + 32) * 32 + d0]; wb.y = W2l[(q + 32) * 32 + d0 + 1]; acc2 = wmma_f32_4(a, wb, acc2);
    wb.x = W2l[(q + 48) * 32 + d0]; wb.y = W2l[(q + 48) * 32 + d0 + 1]; acc3 = wmma_f32_4(a, wb, acc3);
  }
#pragma unroll
  for (int kk = 0; kk < 8; ++kk) {        // h1 @ W2r^T, K=32
    int d0 = kk * 4 + 2 * kh;
    v2f a; a.x = Hrow[d0]; a.y = Hrow[d0 + 1];
    v2f wb;
    wb.x = W2r[q * 32 + d0];        wb.y = W2r[q * 32 + d0 + 1];        acc0 = wmma_f32_4(a, wb, acc0);
    wb.x = W2r[(q + 16) * 32 + d0]; wb.y = W2r[(q + 16) * 32 + d0 + 1]; acc1 = wmma_f32_4(a, wb, acc1);
    wb.x = W2r[(q + 32) * 32 + d0]; wb.y = W2r[(q + 32) * 32 + d0 + 1]; acc2 = wmma_f32_4(a, wb, acc2);
    wb.x = W2r[(q + 48) * 32 + d0]; wb.y = W2r[(q + 48) * 32 + d0 + 1]; acc3 = wmma_f32_4(a, wb, acc3);
  }
#pragma unroll
  for (int r = 0; r < 8; ++r) {
    float v0 = acc0[r] + bias2[q];
    float v1 = acc1[r] + bias2[16 + q];
    float v2 = acc2[r] + bias2[32 + q];
    float v3 = acc3[r] + bias2[48 + q];
    float ss = v0 * v0 + v1 * v1 + v2 * v2 + v3 * v3;
    // reduce across the 16 lanes holding this C row (masks < 16 stay in half-wave)
    ss += __shfl_xor(ss, 1, 32);
    ss += __shfl_xor(ss, 2, 32);
    ss += __shfl_xor(ss, 4, 32);
    ss += __shfl_xor(ss, 8, 32);
    float sc = 1.0f / fmaxf(sqrtf(ss), 1e-12f);
    int row = base + r + 8 * kh;
    gcn[(size_t)row * 64 + q]      = v0 * sc;
    gcn[(size_t)row * 64 + 16 + q] = v1 * sc;
    gcn[(size_t)row * 64 + 32 + q] = v2 * sc;
    gcn[(size_t)row * 64 + 48 + q] = v3 * sc;
  }
}

// ---------------- fused sampling step k: one block (64 threads) per user ----------------
__global__ void kg_step_kernel(const float* __restrict__ gcn, const float* __restrict__ disc,
                               const int* __restrict__ users, const int* __restrict__ adj,
                               int* __restrict__ p_cur, float* __restrict__ out, int k) {
  __shared__ float s_u[64], s_pe[64], s_p[64];
  __shared__ int s_cand[32], s_pos2[32];
  __shared__ float s_logit[32], s_r[32];
  int b = blockIdx.x;
  int m = threadIdx.x;                    // 0..63 = neighbor slot
  int u = users[b];
  int p0 = p_cur[b];
  s_u[m] = gcn[(size_t)u * 64 + m];
  s_pe[m] = gcn[(size_t)p0 * 64 + m];
  __syncthreads();

  // ---- step A: kg(p0, explore=1, step=1): argmax over adj[p0] ----
  int idxA = adj[(size_t)p0 * MM + m];
  float acc = 0.f;
  for (int d = 0; d < 64; ++d) acc += leaky(s_pe[d] * gcn[(size_t)idxA * 64 + d]) * s_u[d];
  s_p[m] = acc;
  __syncthreads();
  float mx = -3.4e38f; int best = 0;
  for (int j = 0; j < 64; ++j) { float v = s_p[j]; if (v > mx) { mx = v; best = j; } }
  float sm = 0.f;
  for (int j = 0; j < 64; ++j) sm += __expf(s_p[j] - mx);
  int one_hop = adj[(size_t)p0 * MM + best];
  float ohl = s_p[best] - mx - __logf(sm);
  __syncthreads();

  // ---- step B: kg(one_hop, explore=1, step=2): top-32 over adj[one_hop] ----
  s_pe[m] = gcn[(size_t)one_hop * 64 + m];
  __syncthreads();
  int idxB = adj[(size_t)one_hop * MM + m];
  acc = 0.f;
  for (int d = 0; d < 64; ++d) acc += leaky(s_pe[d] * gcn[(size_t)idxB * 64 + d]) * s_u[d];
  s_p[m] = acc;
  __syncthreads();
  mx = -3.4e38f;
  for (int j = 0; j < 64; ++j) mx = fmaxf(mx, s_p[j]);
  sm = 0.f;
  for (int j = 0; j < 64; ++j) sm += __expf(s_p[j] - mx);
  float myp = s_p[m];
  int rank = 0;
  for (int j = 0; j < 64; ++j) rank += (s_p[j] > myp) || (s_p[j] == myp && j < m);
  if (rank < 32) {
    int c = idxB;
    if (c < 0 || c > ITEM_HI) c = (int)(hash3(2u * k, (unsigned)b, (unsigned)rank) % (ITEM_HI + 1u));
    s_cand[rank] = c;
    s_logit[rank] = myp - mx - __logf(sm);
  }
  __syncthreads();

  // ---- step C: kg(p0, explore=2, step=2): top-32 over adj[u] (pos2) ----
  s_pe[m] = gcn[(size_t)p0 * 64 + m];
  __syncthreads();
  int idxC = adj[(size_t)u * MM + m];
  acc = 0.f;
  for (int d = 0; d < 64; ++d) acc += leaky(s_pe[d] * gcn[(size_t)idxC * 64 + d]) * s_u[d];
  s_p[m] = acc;
  __syncthreads();
  myp = s_p[m];
  rank = 0;
  for (int j = 0; j < 64; ++j) rank += (s_p[j] > myp) || (s_p[j] == myp && j < m);
  if (rank < 32) {
    int c = idxC;
    if (c < 0 || c > ITEM_HI) c = (int)(hash3(2u * k + 1u, (unsigned)b, (unsigned)rank) % (ITEM_HI + 1u));
    s_pos2[rank] = c;
  }
  __syncthreads();

  // ---- prune: argmin_j  disc[u] . (disc[pos2_j] - disc[cand_j]) ----
  s_u[m] = disc[(size_t)u * 64 + m];
  __syncthreads();
  if (m < 32) {
    int pj = s_pos2[m], nj = s_cand[m];
    float r = 0.f;
    for (int d = 0; d < 64; ++d)
      r += s_u[d] * (disc[(size_t)pj * 64 + d] - disc[(size_t)nj * 64 + d]);
    s_r[m] = r;
  }
  __syncthreads();
  if (m == 0) {
    float mn = s_r[0]; int bi = 0;
    for (int j = 1; j < 32; ++j) if (s_r[j] < mn) { mn = s_r[j]; bi = j; }
    int gn = s_cand[bi];
    out[k * BB + b] = (float)gn;                 // stack(neg_list)[k, b]
    out[2 * BB + k * BB + b] = s_logit[bi] + ohl; // stack(prob_list)[k, b]
    p_cur[b] = gn;
  }
}

extern "C" void kernel_launch(void* const* d_in, const int* in_sizes, int n_in,
                              void* d_out, int out_size, void* d_ws, size_t ws_size,
                              hipStream_t stream) {
  const float* x    = (const float*)d_in[0];   // entity_embedding [N,64]
  const float* W1l  = (const float*)d_in[1];   // [32,64]
  const float* W1r  = (const float*)d_in[2];   // [32,64]
  const float* b1   = (const float*)d_in[3];   // [32]
  const float* W2l  = (const float*)d_in[4];   // [64,32]
  const float* W2r  = (const float*)d_in[5];   // [64,32]
  const float* b2   = (const float*)d_in[6];   // [64]
  const float* disc = (const float*)d_in[7];   // disc_emb [N,64]
  const int* users  = (const int*)d_in[8];     // [B]
  const int* pos    = (const int*)d_in[9];     // [B]
  const int* adj    = (const int*)d_in[10];    // [N,64]
  const int* edge   = (const int*)d_in[11];    // [N,32]
  float* out = (float*)d_out;

  char* ws = (char*)d_ws;
  size_t off = 0;
  float* sum1 = (float*)(ws + off); off += (size_t)NN * 64 * sizeof(float);
  float* cnt  = (float*)(ws + off); off += (size_t)NN * sizeof(float);
  float* sum2 = (float*)(ws + off); off += (size_t)NN * 32 * sizeof(float);
  float* h1   = (float*)(ws + off); off += (size_t)NN * 32 * sizeof(float);
  float* gcn  = (float*)(ws + off); off += (size_t)NN * 64 * sizeof(float);
  int* p_cur  = (int*)(ws + off);   off += (size_t)BB * sizeof(int);
  (void)ws_size; (void)in_sizes; (void)n_in; (void)out_size;

  // sum1, cnt, sum2 are contiguous: zero in one pass
  long nz = (long)NN * 64 + NN + (long)NN * 32;
  zero_f32_kernel<<<2048, 256, 0, stream>>>(sum1, nz);
  init_p_kernel<<<(BB + 255) / 256, 256, 0, stream>>>(pos, p_cur);

  scatter1_kernel<<<(NN * 32 + 255) / 256, 256, 0, stream>>>(x, edge, sum1, cnt);

  int nt = NN / 16;                         // 6250 tiles, 4 waves per 128-thread block
  int gemmBlocks = (nt + 3) / 4;
  gemm1_kernel<<<gemmBlocks, 128, 0, stream>>>(sum1, cnt, x, W1l, W1r, b1, h1);

  scatter2_kernel<<<(NN * 32 + 255) / 256, 256, 0, stream>>>(h1, edge, sum2);

  gemm2_kernel<<<gemmBlocks, 128, 0, stream>>>(sum2, cnt, h1, W2l, W2r, b2, gcn);

  kg_step_kernel<<<BB, 64, 0, stream>>>(gcn, disc, users, adj, p_cur, out, 0);
  kg_step_kernel<<<BB, 64, 0, stream>>>(gcn, disc, users, adj, p_cur, out, 1);
}